// BiGraphContrastLayer_8108898255226
// MI455X (gfx1250) — compile-verified
//
#include <hip/hip_runtime.h>
#include <math.h>

typedef __attribute__((ext_vector_type(2))) float v2f;
typedef __attribute__((ext_vector_type(8))) float v8f;

#define N_SRC 32768
#define N_DST 32768
#define N_TOT (N_SRC + N_DST)
#define D 256
#define EPS 1e-8f

// ---------------------------------------------------------------------------
// Wave (32-lane) sum reduction
// ---------------------------------------------------------------------------
__device__ inline float wave_reduce_sum(float v) {
  #pragma unroll
  for (int off = 16; off > 0; off >>= 1)
    v += __shfl_down(v, off, 32);
  return v;
}

// ---------------------------------------------------------------------------
// GEMM: m[N_TOT x D] = feat[N_TOT x D] @ W[D x D] using V_WMMA_F32_16X16X4_F32.
// One wave -> 16x64 output tile (4 accumulators), K stepped by 4.
// A fragment (16x4 f32): lane%16 = row, lane/16 = K-half, {K, K+1} per lane.
// B fragment (4x16 f32): lane%16 = col, lane/16 = K-half, {K, K+1} per lane.
// D (16x16 f32): lane%16 = col, VGPR r -> row = r + 8*(lane/16).
// ---------------------------------------------------------------------------
__global__ __launch_bounds__(256) void gemm_wmma_kernel(
    const float* __restrict__ feat, const float* __restrict__ W,
    float* __restrict__ m) {
  int wave = (int)((blockIdx.x * blockDim.x + threadIdx.x) >> 5);
  int lane = (int)(threadIdx.x & 31);
  int row_tile = wave >> 2;          // 0..4095 (N_TOT/16)
  int col_grp  = wave & 3;           // 0..3  (each 64 cols)
  if (row_tile >= N_TOT / 16) return;
  int row0 = row_tile * 16;
  int col0 = col_grp * 64;
  int lrow  = lane & 15;
  int khalf = lane >> 4;

  v8f acc0 = {}, acc1 = {}, acc2 = {}, acc3 = {};
  for (int k0 = 0; k0 < D; k0 += 4) {
    int kb = k0 + khalf * 2;
    const float* ap = feat + (size_t)(row0 + lrow) * D + kb;
    v2f a; a.x = ap[0]; a.y = ap[1];
    const float* bp = W + (size_t)kb * D + col0 + lrow;
    v2f b0; b0.x = bp[0];  b0.y = bp[0 + D];
    v2f b1; b1.x = bp[16]; b1.y = bp[16 + D];
    v2f b2; b2.x = bp[32]; b2.y = bp[32 + D];
    v2f b3; b3.x = bp[48]; b3.y = bp[48 + D];
    acc0 = __builtin_amdgcn_wmma_f32_16x16x4_f32(false, a, false, b0, (short)0, acc0, false, false);
    acc1 = __builtin_amdgcn_wmma_f32_16x16x4_f32(false, a, false, b1, (short)0, acc1, false, false);
    acc2 = __builtin_amdgcn_wmma_f32_16x16x4_f32(false, a, false, b2, (short)0, acc2, false, false);
    acc3 = __builtin_amdgcn_wmma_f32_16x16x4_f32(false, a, false, b3, (short)0, acc3, false, false);
  }
  float* out = m + (size_t)(row0 + khalf * 8) * D + col0 + lrow;
  #pragma unroll
  for (int r = 0; r < 8; ++r) {
    out[(size_t)r * D + 0]  = acc0[r];
    out[(size_t)r * D + 16] = acc1[r];
    out[(size_t)r * D + 32] = acc2[r];
    out[(size_t)r * D + 48] = acc3[r];
  }
}

// ---------------------------------------------------------------------------
// Self-loop init: agg_pos = agg_neg = m (elementwise over N_TOT*D)
// ---------------------------------------------------------------------------
__global__ void init_copy_kernel(const float* __restrict__ m,
                                 float* __restrict__ hp, float* __restrict__ hn,
                                 long n) {
  long i = (long)blockIdx.x * blockDim.x + threadIdx.x;
  if (i < n) { float v = m[i]; hp[i] = v; hn[i] = v; }
}

// deg = 1 (self loop), zero global accumulators
__global__ void init_small_kernel(float* __restrict__ degp, float* __restrict__ degn,
                                  float* __restrict__ accum) {
  int i = (int)(blockIdx.x * blockDim.x + threadIdx.x);
  if (i < N_TOT) { degp[i] = 1.0f; degn[i] = 1.0f; }
  if (i < 2) accum[i] = 0.0f;
}

// ---------------------------------------------------------------------------
// Edge scatter: agg[dst] += m[src] (256 floats), deg[dst] += 1.  Wave per edge.
// Prefetch the gathered source row before the reads become demand misses.
// ---------------------------------------------------------------------------
__global__ __launch_bounds__(256) void scatter_add_kernel(
    const int* __restrict__ src, const int* __restrict__ dst, int ne,
    const float* __restrict__ m, float* __restrict__ agg, float* __restrict__ deg) {
  int e    = (int)((blockIdx.x * blockDim.x + threadIdx.x) >> 5);
  int lane = (int)(threadIdx.x & 31);
  if (e >= ne) return;
  int s = src[e], d = dst[e];
  const float* mr = m + (size_t)s * D + lane * 8;
  float* ar = agg + (size_t)d * D + lane * 8;
  __builtin_prefetch(mr, 0, 3);               // global_prefetch_b8 (read)
  float4 x0 = *(const float4*)(mr);
  float4 x1 = *(const float4*)(mr + 4);
  atomicAdd(ar + 0, x0.x); atomicAdd(ar + 1, x0.y);
  atomicAdd(ar + 2, x0.z); atomicAdd(ar + 3, x0.w);
  atomicAdd(ar + 4, x1.x); atomicAdd(ar + 5, x1.y);
  atomicAdd(ar + 6, x1.z); atomicAdd(ar + 7, x1.w);
  if (lane == 0) atomicAdd(deg + d, 1.0f);
}

// ---------------------------------------------------------------------------
// Finalize conv: h = agg/max(deg,1) + b, PReLU; per-row inverse norm. Wave/row.
// ---------------------------------------------------------------------------
__global__ __launch_bounds__(256) void finalize_kernel(
    float* __restrict__ h, const float* __restrict__ deg,
    const float* __restrict__ b, const float* __restrict__ prelu_a,
    float* __restrict__ invn) {
  int v    = (int)((blockIdx.x * blockDim.x + threadIdx.x) >> 5);
  int lane = (int)(threadIdx.x & 31);
  if (v >= N_TOT) return;
  float dg = fmaxf(deg[v], 1.0f);
  float slope = prelu_a[0];
  float* row = h + (size_t)v * D + lane * 8;
  const float* bp = b + lane * 8;
  float vals[8];
  float ss = 0.0f;
  #pragma unroll
  for (int j = 0; j < 8; ++j) {
    float x = row[j] / dg + bp[j];
    x = (x >= 0.0f) ? x : slope * x;
    vals[j] = x;
    ss += x * x;
  }
  ss = wave_reduce_sum(ss);
  #pragma unroll
  for (int j = 0; j < 8; ++j) row[j] = vals[j];
  if (lane == 0) invn[v] = 1.0f / fmaxf(sqrtf(ss), EPS);
}

// ---------------------------------------------------------------------------
// Positive pairs (wave per predict node): p = cos(hp[v], hn[v]).
// Also seeds neg0/neg1 with the self-loop cosine and accumulates exp/sum.
// ---------------------------------------------------------------------------
__global__ __launch_bounds__(256) void pos_loss_kernel(
    const float* __restrict__ hp, const float* __restrict__ hn,
    const float* __restrict__ invp, const float* __restrict__ invq,
    float* __restrict__ posl, float* __restrict__ neg0, float* __restrict__ neg1,
    float* __restrict__ accum) {
  int v    = (int)((blockIdx.x * blockDim.x + threadIdx.x) >> 5);
  int lane = (int)(threadIdx.x & 31);
  if (v >= N_DST) return;
  int node = N_SRC + v;
  const float* ar = hp + (size_t)node * D + lane * 8;
  const float* br = hn + (size_t)node * D + lane * 8;
  float4 a0 = *(const float4*)(ar),     b0 = *(const float4*)(br);
  float4 a1 = *(const float4*)(ar + 4), b1 = *(const float4*)(br + 4);
  float s = a0.x*b0.x + a0.y*b0.y + a0.z*b0.z + a0.w*b0.w
          + a1.x*b1.x + a1.y*b1.y + a1.z*b1.z + a1.w*b1.w;
  s = wave_reduce_sum(s);
  if (lane == 0) {
    float p = s * invp[node] * invq[node];
    posl[v] = p;
    neg0[v] = p;   // self-loop contribution to _seg_cos(hp, hn, ...)
    neg1[v] = p;   // self-loop contribution to _seg_cos(hn, hp, ...)
    atomicAdd(accum + 0, expf(p));
    atomicAdd(accum + 1, p);
  }
}

// ---------------------------------------------------------------------------
// Per-edge cosine scatter: out[d-N_SRC] += cos(anchor[d], nei[s]). Wave/edge.
// ---------------------------------------------------------------------------
__global__ __launch_bounds__(256) void edge_cos_kernel(
    const int* __restrict__ src, const int* __restrict__ dst, int ne,
    const float* __restrict__ anchor, const float* __restrict__ invA,
    const float* __restrict__ nei, const float* __restrict__ invB,
    float* __restrict__ out) {
  int e    = (int)((blockIdx.x * blockDim.x + threadIdx.x) >> 5);
  int lane = (int)(threadIdx.x & 31);
  if (e >= ne) return;
  int s = src[e], d = dst[e];
  if (d < N_SRC) return;  // reference masks to predict partition
  const float* ar = anchor + (size_t)d * D + lane * 8;
  const float* br = nei + (size_t)s * D + lane * 8;
  __builtin_prefetch(ar, 0, 3);               // global_prefetch_b8
  __builtin_prefetch(br, 0, 3);
  float4 a0 = *(const float4*)(ar),     b0 = *(const float4*)(br);
  float4 a1 = *(const float4*)(ar + 4), b1 = *(const float4*)(br + 4);
  float sum = a0.x*b0.x + a0.y*b0.y + a0.z*b0.z + a0.w*b0.w
            + a1.x*b1.x + a1.y*b1.y + a1.z*b1.z + a1.w*b1.w;
  sum = wave_reduce_sum(sum);
  if (lane == 0) atomicAdd(out + (d - N_SRC), sum * invA[d] * invB[s]);
}

// ---------------------------------------------------------------------------
// sum(exp(neg[i])) into accum[0]
// ---------------------------------------------------------------------------
__global__ __launch_bounds__(256) void exp_reduce_kernel(
    const float* __restrict__ neg, int n, float* __restrict__ accum) {
  __shared__ float smem[256];
  int i = (int)(blockIdx.x * blockDim.x + threadIdx.x);
  float v = (i < n) ? expf(neg[i]) : 0.0f;
  smem[threadIdx.x] = v;
  __syncthreads();
  for (int s = 128; s > 0; s >>= 1) {
    if ((int)threadIdx.x < s) smem[threadIdx.x] += smem[threadIdx.x + s];
    __syncthreads();
  }
  if (threadIdx.x == 0) atomicAdd(accum + 0, smem[0]);
}

__global__ void loss_kernel(const float* __restrict__ accum, float* __restrict__ out) {
  if (blockIdx.x == 0 && threadIdx.x == 0) out[0] = logf(accum[0]) - accum[1];
}

// d_out[1 + i] = h_pos[N_SRC:][i]
__global__ void copy_out_kernel(const float* __restrict__ hp, float* __restrict__ out, long n) {
  long i = (long)blockIdx.x * blockDim.x + threadIdx.x;
  if (i < n) out[1 + i] = hp[(size_t)N_SRC * D + i];
}

// ---------------------------------------------------------------------------
extern "C" void kernel_launch(void* const* d_in, const int* in_sizes, int n_in,
                              void* d_out, int out_size, void* d_ws, size_t ws_size,
                              hipStream_t stream) {
  const float* feat    = (const float*)d_in[0];
  const int*   src     = (const int*)d_in[1];
  const int*   dst     = (const int*)d_in[2];
  const int*   neg_src = (const int*)d_in[3];
  const int*   neg_dst = (const int*)d_in[4];
  const float* W       = (const float*)d_in[5];
  const float* b       = (const float*)d_in[6];
  const float* prelu_a = (const float*)d_in[7];
  float* out = (float*)d_out;

  const int E     = in_sizes[1];
  const int E_NEG = in_sizes[3];

  const size_t nd = (size_t)N_TOT * D;
  float* ws   = (float*)d_ws;
  float* m    = ws;               // nd
  float* hp   = m + nd;           // nd  (agg_pos -> h_pos)
  float* hn   = hp + nd;          // nd  (agg_neg -> h_neg)
  float* degp = hn + nd;          // N_TOT
  float* degn = degp + N_TOT;     // N_TOT
  float* invp = degn + N_TOT;     // N_TOT
  float* invq = invp + N_TOT;     // N_TOT
  float* posl = invq + N_TOT;     // N_DST
  float* neg0 = posl + N_DST;     // N_DST
  float* neg1 = neg0 + N_DST;     // N_DST (contiguous with neg0)
  float* accum = neg1 + N_DST;    // 2

  // 1) GEMM m = feat @ W (WMMA f32): 16384 waves of 16x64 tiles
  gemm_wmma_kernel<<<(N_TOT / 16) * 4 * 32 / 256, 256, 0, stream>>>(feat, W, m);

  // 2) Self-loop init + degrees + accumulators
  init_copy_kernel<<<(int)((nd + 255) / 256), 256, 0, stream>>>(m, hp, hn, (long)nd);
  init_small_kernel<<<(N_TOT + 255) / 256, 256, 0, stream>>>(degp, degn, accum);

  // 3) Edge aggregation (wave per edge)
  scatter_add_kernel<<<(E * 32 + 255) / 256, 256, 0, stream>>>(src, dst, E, m, hp, degp);
  scatter_add_kernel<<<(E_NEG * 32 + 255) / 256, 256, 0, stream>>>(neg_src, neg_dst, E_NEG, m, hn, degn);

  // 4) Normalize by degree, bias, PReLU, row inverse-norm
  finalize_kernel<<<(N_TOT * 32 + 255) / 256, 256, 0, stream>>>(hp, degp, b, prelu_a, invp);
  finalize_kernel<<<(N_TOT * 32 + 255) / 256, 256, 0, stream>>>(hn, degn, b, prelu_a, invq);

  // 5) Positive cosines (also self-loop seeds for the two seg_cos terms)
  pos_loss_kernel<<<(N_DST * 32 + 255) / 256, 256, 0, stream>>>(
      hp, hn, invp, invq, posl, neg0, neg1, accum);

  // 6) Per-edge cosines: hp anchors vs neg graph, hn anchors vs pos graph
  edge_cos_kernel<<<(E_NEG * 32 + 255) / 256, 256, 0, stream>>>(
      neg_src, neg_dst, E_NEG, hp, invp, hn, invq, neg0);
  edge_cos_kernel<<<(E * 32 + 255) / 256, 256, 0, stream>>>(
      src, dst, E, hn, invq, hp, invp, neg1);

  // 7) logsumexp accumulation over neg0|neg1 (contiguous 2*N_DST), final loss
  exp_reduce_kernel<<<(2 * N_DST + 255) / 256, 256, 0, stream>>>(neg0, 2 * N_DST, accum);
  loss_kernel<<<1, 1, 0, stream>>>(accum, out);

  // 8) h_pos[N_SRC:] -> d_out[1:]
  copy_out_kernel<<<(int)(((size_t)N_DST * D + 255) / 256), 256, 0, stream>>>(
      hp, out, (long)N_DST * D);
}